// GCN_84207128805537
// MI455X (gfx1250) — compile-verified
//
#include <hip/hip_runtime.h>

#define NN   100000
#define NE   3200000
#define HID  128
#define NCLS 10
#define NG   128
#define EPSV 1e-5f

typedef float v2f __attribute__((ext_vector_type(2)));
typedef float v8f __attribute__((ext_vector_type(8)));
typedef unsigned int v4u __attribute__((ext_vector_type(4)));
typedef int v4i __attribute__((ext_vector_type(4)));
typedef int v8i __attribute__((ext_vector_type(8)));

// ---------------- workspace layout (floats) ----------------
#define OFF_AGGX  100000
#define OFF_STATS 200000
#define OFF_HP1   200512
#define OFF_HPA   216896
#define OFF_CNT   233280
#define OFF_H1    233408
#define OFF_AGGH  13033408
#define ZERO_CNT  33408   // floats to zero starting at OFF_STATS

#define LDSW 132          // padded LDS row stride (128 + 4 dwords TDM padding)

// ---------------- init: deg=1 (self loop), zero stats/pools ----------------
__global__ void k_init(float* __restrict__ ws) {
    int i = blockIdx.x * blockDim.x + threadIdx.x;
    if (i < NN) ws[i] = 1.0f;
    if (i < ZERO_CNT) ws[OFF_STATS + i] = 0.0f;
}

__global__ void k_deg(float* __restrict__ deg, const int* __restrict__ dst) {
    int e = blockIdx.x * blockDim.x + threadIdx.x;
    if (e < NE) atomicAdd(&deg[dst[e]], 1.0f);
}

__global__ void k_dinv(float* __restrict__ ws, const float* __restrict__ x) {
    int i = blockIdx.x * blockDim.x + threadIdx.x;
    if (i < NN) {
        float d = __frsqrt_rn(ws[i]);
        ws[i] = d;
        ws[OFF_AGGX + i] = x[i] * d * d;
    }
}

__global__ void k_aggx(float* __restrict__ ws, const float* __restrict__ x,
                       const int* __restrict__ src, const int* __restrict__ dst) {
    int e = blockIdx.x * blockDim.x + threadIdx.x;
    if (e < NE) {
        int s = src[e], d = dst[e];
        atomicAdd(&ws[OFF_AGGX + d], x[s] * ws[s] * ws[d]);
    }
}

// ------- layer 1: h1raw = relu(aggx*Wc1 + x*Wl1 + b), accumulate BN1 stats --
__global__ void k_l1(float* __restrict__ ws, const float* __restrict__ x,
                     const float* __restrict__ Wc1, const float* __restrict__ bc1,
                     const float* __restrict__ Wl1, const float* __restrict__ bl1) {
    const int f = threadIdx.x;
    const int b = blockIdx.x;
    float wc = Wc1[f], wl = Wl1[f], bb = bc1[f] + bl1[f];
    float s = 0.0f, s2 = 0.0f;
    float* h1 = ws + OFF_H1;
    #pragma unroll 4
    for (int r = 0; r < 16; ++r) {
        int i = b * 16 + r;
        float v = ws[OFF_AGGX + i] * wc + x[i] * wl + bb;
        v = fmaxf(v, 0.0f);
        h1[i * HID + f] = v;
        s += v; s2 += v * v;
    }
    atomicAdd(&ws[OFF_STATS + f],       s);
    atomicAdd(&ws[OFF_STATS + 128 + f], s2);
}

// ------- apply BN1 in place; init agg_h with self-loop term ----------------
__global__ void k_bn1(float* __restrict__ ws, const float* __restrict__ g1,
                      const float* __restrict__ be1) {
    int idx = blockIdx.x * blockDim.x + threadIdx.x;
    if (idx >= NN * HID) return;
    int i = idx >> 7, f = idx & 127;
    float mu  = ws[OFF_STATS + f] * (1.0f / NN);
    float var = fmaxf(ws[OFF_STATS + 128 + f] * (1.0f / NN) - mu * mu, 0.0f);
    float inv = __frsqrt_rn(var + EPSV);
    float v = (ws[OFF_H1 + idx] - mu) * inv * g1[f] + be1[f];
    ws[OFF_H1 + idx] = v;
    float dn = ws[i];
    ws[OFF_AGGH + idx] = v * dn * dn;
}

// ------- layer-2 edge scatter: one wave per edge, 4 feats/lane -------------
__global__ void k_conv2(float* __restrict__ ws, const int* __restrict__ src,
                        const int* __restrict__ dst) {
    long long gid = (long long)blockIdx.x * blockDim.x + threadIdx.x;
    if (gid >= (long long)NE * 32) return;
    int e = (int)(gid >> 5);
    int f = (int)(gid & 31) << 2;
    int s = src[e], d = dst[e];
    float nrm = ws[s] * ws[d];
    const float4 hv = *(const float4*)(ws + OFF_H1 + (long long)s * HID + f);
    float* out = ws + OFF_AGGH + (long long)d * HID + f;
    atomicAdd(out + 0, hv.x * nrm);
    atomicAdd(out + 1, hv.y * nrm);
    atomicAdd(out + 2, hv.z * nrm);
    atomicAdd(out + 3, hv.w * nrm);
}

// ------- per-graph pooling of h1 and agg_h ---------------------------------
__global__ void k_pool(float* __restrict__ ws, const int* __restrict__ batch) {
    int idx = blockIdx.x * blockDim.x + threadIdx.x;
    if (idx >= NN * HID) return;
    int i = idx >> 7, f = idx & 127;
    int g = batch[i];
    atomicAdd(&ws[OFF_HP1 + g * HID + f], ws[OFF_H1 + idx]);
    atomicAdd(&ws[OFF_HPA + g * HID + f], ws[OFF_AGGH + idx]);
    if (f == 0) atomicAdd(&ws[OFF_CNT + g], 1.0f);
}

// ------- TDM descriptor helper: 2D tile [16 rows x 128 f32] -> padded LDS --
__device__ __forceinline__ void tdm_load_tile(const float* gsrc, unsigned int lds_byte) {
    unsigned long long ga = (unsigned long long)(size_t)gsrc;
    v4u g0 = { 1u,                               // count=1, all flags 0
               lds_byte,                         // lds_addr
               (unsigned int)ga,                 // global_addr[31:0]
               (unsigned int)((ga >> 32) & 0x1FFFFFFu) | (2u << 30) }; // [56:32] | type=2
    v8i g1 = { (int)((2u << 16) | (1u << 20) | (6u << 22) | (3u << 25)),
               // data_size=4B, pad_enable, pad_interval=128dw, pad_amount=4dw
               (int)(128u << 16),                // tensor_dim0 = 128
               (int)(16u  << 16),                // tensor_dim1 = 16
               (int)(128u << 16),                // tile_dim0 = 128
               16,                               // tile_dim1 = 16, tile_dim2 = 0
               128,                              // tensor_dim0_stride = 128
               0, 0 };
    v4i z4 = { 0, 0, 0, 0 };
#if __clang_major__ >= 23
    v8i z8 = { 0, 0, 0, 0, 0, 0, 0, 0 };
    __builtin_amdgcn_tensor_load_to_lds(g0, g1, z4, z4, z8, 0);
#else
    __builtin_amdgcn_tensor_load_to_lds(g0, g1, z4, z4, 0);
#endif
}

// ------- layer-2 GEMM: TDM-staged A tiles in LDS + WMMA f32 16x16x4 --------
// One block = one 16-row tile; 8 waves cover all 128 output columns.
// D = agg_h @ Wc2 + h1 @ Wl2 + (bc2+bl2); only BN2 sum/sumsq leave the core.
__global__ void k_gemm2(const float* __restrict__ ws, const float* __restrict__ Wc2,
                        const float* __restrict__ Wl2, const float* __restrict__ bc2,
                        const float* __restrict__ bl2, float* __restrict__ stats) {
    __shared__ float As[16 * LDSW];   // agg_h tile (TDM-padded rows)
    __shared__ float Ah[16 * LDSW];   // h1 tile
    const int wave = threadIdx.x >> 5;
    const int lane = threadIdx.x & 31;
    const int r0   = blockIdx.x * 16;

    if (wave == 0) {
        tdm_load_tile(ws + OFF_AGGH + (long long)r0 * HID, (unsigned int)(size_t)As);
        tdm_load_tile(ws + OFF_H1   + (long long)r0 * HID, (unsigned int)(size_t)Ah);
        __builtin_amdgcn_s_wait_tensorcnt(0);
    }
    __syncthreads();

    const int col0 = wave * 16;
    const int m    = lane & 15;
    const int koff = (lane >> 4) << 1;             // 0 or 2
    const float* Ar0 = As + m * LDSW;              // ds_load_b64, bank-conflict-free
    const float* Ar1 = Ah + m * LDSW;
    v8f c = {};
    #pragma unroll 8
    for (int kb = 0; kb < HID; kb += 4) {
        int k0 = kb + koff;
        v2f a0 = *(const v2f*)(Ar0 + k0);
        v2f b0; b0.x = Wc2[k0 * HID + col0 + m]; b0.y = Wc2[(k0 + 1) * HID + col0 + m];
        c = __builtin_amdgcn_wmma_f32_16x16x4_f32(false, a0, false, b0, (short)0, c, false, false);
        v2f a1 = *(const v2f*)(Ar1 + k0);
        v2f b1; b1.x = Wl2[k0 * HID + col0 + m]; b1.y = Wl2[(k0 + 1) * HID + col0 + m];
        c = __builtin_amdgcn_wmma_f32_16x16x4_f32(false, a1, false, b1, (short)0, c, false, false);
    }
    float bb = bc2[col0 + m] + bl2[col0 + m];
    float s = 0.0f, s2 = 0.0f;
    #pragma unroll
    for (int j = 0; j < 8; ++j) {
        float v = c[j] + bb;
        s += v; s2 += v * v;
    }
    s  += __shfl_xor(s, 16, 32);
    s2 += __shfl_xor(s2, 16, 32);
    if (lane < 16) {
        atomicAdd(&stats[256 + col0 + lane], s);
        atomicAdd(&stats[384 + col0 + lane], s2);
    }
}

// ------- finale: pooled pre-acts via small GEMM, BN2 affine, classifier ----
__global__ void k_final(const float* __restrict__ ws, const float* __restrict__ Wc2,
                        const float* __restrict__ Wl2, const float* __restrict__ bc2,
                        const float* __restrict__ bl2, const float* __restrict__ g2,
                        const float* __restrict__ be2, const float* __restrict__ W3,
                        const float* __restrict__ b3, float* __restrict__ out) {
    int g = threadIdx.x;
    float cntg = ws[OFF_CNT + g];
    float rinv = 1.0f / fmaxf(cntg, 1.0f);
    float acc[NCLS];
    #pragma unroll
    for (int cc = 0; cc < NCLS; ++cc) acc[cc] = b3[cc];
    for (int f = 0; f < HID; ++f) {
        float pre = cntg * (bc2[f] + bl2[f]);
        for (int k = 0; k < HID; ++k) {
            pre += ws[OFF_HPA + g * HID + k] * Wc2[k * HID + f];
            pre += ws[OFF_HP1 + g * HID + k] * Wl2[k * HID + f];
        }
        float mean = pre * rinv;
        float mu  = ws[OFF_STATS + 256 + f] * (1.0f / NN);
        float var = fmaxf(ws[OFF_STATS + 384 + f] * (1.0f / NN) - mu * mu, 0.0f);
        float hb  = (mean - mu) * __frsqrt_rn(var + EPSV) * g2[f] + be2[f];
        #pragma unroll
        for (int cc = 0; cc < NCLS; ++cc) acc[cc] += hb * W3[f * NCLS + cc];
    }
    for (int cc = 0; cc < NCLS; ++cc) out[g * NCLS + cc] = acc[cc];
}

extern "C" void kernel_launch(void* const* d_in, const int* in_sizes, int n_in,
                              void* d_out, int out_size, void* d_ws, size_t ws_size,
                              hipStream_t stream) {
    const float* x     = (const float*)d_in[0];
    const int*   ei    = (const int*)d_in[1];
    const int*   batch = (const int*)d_in[2];
    const float* Wc1   = (const float*)d_in[3];
    const float* bc1   = (const float*)d_in[4];
    const float* Wl1   = (const float*)d_in[5];
    const float* bl1   = (const float*)d_in[6];
    const float* g1    = (const float*)d_in[7];
    const float* be1   = (const float*)d_in[8];
    const float* Wc2   = (const float*)d_in[9];
    const float* bc2   = (const float*)d_in[10];
    const float* Wl2   = (const float*)d_in[11];
    const float* bl2   = (const float*)d_in[12];
    const float* g2    = (const float*)d_in[13];
    const float* be2   = (const float*)d_in[14];
    const float* W3    = (const float*)d_in[15];
    const float* b3    = (const float*)d_in[16];
    float* ws  = (float*)d_ws;
    float* out = (float*)d_out;
    const int* src = ei;
    const int* dst = ei + NE;

    k_init <<<(NN + 255) / 256, 256, 0, stream>>>(ws);
    k_deg  <<<(NE + 255) / 256, 256, 0, stream>>>(ws, dst);
    k_dinv <<<(NN + 255) / 256, 256, 0, stream>>>(ws, x);
    k_aggx <<<(NE + 255) / 256, 256, 0, stream>>>(ws, x, src, dst);
    k_l1   <<<NN / 16, 128, 0, stream>>>(ws, x, Wc1, bc1, Wl1, bl1);
    k_bn1  <<<(NN * HID) / 256, 256, 0, stream>>>(ws, g1, be1);
    k_conv2<<<(NE * 32) / 256, 256, 0, stream>>>(ws, src, dst);
    k_pool <<<(NN * HID) / 256, 256, 0, stream>>>(ws, batch);
    k_gemm2<<<NN / 16, 256, 0, stream>>>(ws, Wc2, Wl2, bc2, bl2, ws + OFF_STATS);
    k_final<<<1, NG, 0, stream>>>(ws, Wc2, Wl2, bc2, bl2, g2, be2, W3, b3, out);
}